// Sim_53377853555121
// MI455X (gfx1250) — compile-verified
//
#include <hip/hip_runtime.h>

// CDNA5 / gfx1250: batched cosine-similarity MSE via Gram-matrix trick.
//   mean((Xn Xn^T - 1)^2) = (||G||_F^2 - 2*||colsum||^2 + B*N^2) / (B*N^2)
// with G = Xn^T Xn (128x128 per batch) computed with V_WMMA_F32_16X16X4_F32.
//
// LDS holds normalized rows in row-PAIR-interleaved form:
//   element (row, col) -> lds[(row/2)*PSTR + 2*col + (row&1)]
// so a WMMA A/B fragment {Xn[k][c], Xn[k+1][c]} is one aligned ds_load_b64
// straight into an even VGPR pair (no register reassembly before v_wmma).

typedef float v2f __attribute__((ext_vector_type(2)));
typedef float v8f __attribute__((ext_vector_type(8)));

#define NBATCH  64
#define NROWS   1024
#define NDIM    128
#define CHUNK   64                 // rows staged in LDS per iteration (32 pairs)
#define PSTR    (2 * NDIM + 32)    // 288 dwords: pair-row offset == 32 mod 64 banks ->
                                   // half-waves hit disjoint bank groups on frag loads

__global__ __launch_bounds__(256)
void gram_batch_kernel(const float* __restrict__ X, float* __restrict__ partials) {
    const int b    = blockIdx.x;          // batch
    const int tid  = threadIdx.x;
    const int w    = tid >> 5;            // wave 0..7  (wave32)
    const int lane = tid & 31;
    const int m    = lane & 15;           // row/col within 16-wide tile
    const int half = lane >> 4;           // selects K pair {0,1} vs {2,3}

    __shared__ float lds[(CHUNK / 2) * PSTR];   // 36,864 B staging
    __shared__ float colbuf[8 * NDIM];          // per-wave column sums
    __shared__ float wavered[8];                // per-wave Frobenius partials

    const float* Xb = X + (size_t)b * NROWS * NDIM;

    v8f acc[8];
    #pragma unroll
    for (int t = 0; t < 8; ++t) acc[t] = {};    // 8 tiles (ti, tj=w) of G

    float csum[4] = {0.f, 0.f, 0.f, 0.f};       // lane owns cols lane+32j, j=0..3

    const int bofs = 32 * w;                    // B-tile dword offset (per-wave constant)

    for (int chunk = 0; chunk < NROWS / CHUNK; ++chunk) {
        const int row0 = chunk * CHUNK;

        // ---- Phase A: load rows (coalesced 128B per b32 load), normalize, stage ----
        #pragma unroll 2
        for (int i = 0; i < CHUNK / 8; ++i) {
            const int lr = w + 8 * i;                  // row within chunk
            const int r  = row0 + lr;                  // row within batch
            const float* rowp = Xb + (size_t)r * NDIM;
            if (chunk + 1 < NROWS / CHUNK)             // stream next chunk into caches
                __builtin_prefetch(rowp + (size_t)CHUNK * NDIM + lane, 0, 0);

            float v[4];
            #pragma unroll
            for (int j = 0; j < 4; ++j) v[j] = rowp[lane + 32 * j];

            float ss = v[0]*v[0] + v[1]*v[1] + v[2]*v[2] + v[3]*v[3];
            #pragma unroll
            for (int off = 16; off > 0; off >>= 1) ss += __shfl_xor(ss, off, 32);

            const float sc = 1.0f / fmaxf(sqrtf(ss), 1e-12f);   // matches F.normalize eps

            // pair-interleaved store: dword (lr/2)*PSTR + 2*col + (lr&1), col = lane+32j
            float* dst = &lds[(lr >> 1) * PSTR + 2 * lane + (lr & 1)];
            #pragma unroll
            for (int j = 0; j < 4; ++j) {
                const float nv = v[j] * sc;
                dst[64 * j] = nv;                      // banks 2*lane+par: conflict-free
                csum[j] += nv;
            }
        }
        __syncthreads();

        // ---- Phase B: G += Xn_chunk^T * Xn_chunk via fp32 WMMA ----
        // Fragment for column-block t, K-step ks: rows 4ks+2*half (+1) at col 16t+m
        // = one b64 at pair (2ks+half), dword 2*(16t+m).
        for (int ks = 0; ks < CHUNK / 4; ++ks) {
            const float* p0 = &lds[(2 * ks + half) * PSTR + 2 * m];

            const v2f bfrag = *reinterpret_cast<const v2f*>(p0 + bofs);

            v2f frag[8];
            #pragma unroll
            for (int t = 0; t < 8; ++t)
                frag[t] = *reinterpret_cast<const v2f*>(p0 + 32 * t);

            #pragma unroll
            for (int t = 0; t < 8; ++t) {
                acc[t] = __builtin_amdgcn_wmma_f32_16x16x4_f32(
                    false, frag[t], false, bfrag, (short)0, acc[t], false, false);
            }
        }
        __syncthreads();
    }

    // ---- per-wave column sums to LDS (fixed slots, no atomics -> deterministic) ----
    #pragma unroll
    for (int j = 0; j < 4; ++j)
        colbuf[w * NDIM + lane + 32 * j] = csum[j];

    // ---- ||G||_F^2 over this wave's 8 tiles ----
    float g2 = 0.0f;
    #pragma unroll
    for (int t = 0; t < 8; ++t) {
        #pragma unroll
        for (int c = 0; c < 8; ++c) { const float g = acc[t][c]; g2 += g * g; }
    }
    #pragma unroll
    for (int off = 16; off > 0; off >>= 1) g2 += __shfl_xor(g2, off, 32);
    if (lane == 0) wavered[w] = g2;
    __syncthreads();

    if (tid == 0) {
        float sumG2 = 0.0f;
        for (int i = 0; i < 8; ++i) sumG2 += wavered[i];
        float sumC2 = 0.0f;
        for (int d = 0; d < NDIM; ++d) {
            float c = 0.0f;
            for (int i = 0; i < 8; ++i) c += colbuf[i * NDIM + d];
            sumC2 += c * c;
        }
        // sum over this batch of (S-1)^2 = sum(S^2) - 2*sum(S) + N^2
        partials[b] = sumG2 - 2.0f * sumC2 + (float)(NROWS) * (float)(NROWS);
    }
}

__global__ void finalize_kernel(const float* __restrict__ partials,
                                float* __restrict__ out) {
    float s = 0.0f;
    for (int i = 0; i < NBATCH; ++i) s += partials[i];   // fixed order, deterministic
    out[0] = s / ((float)NBATCH * (float)NROWS * (float)NROWS);
}

extern "C" void kernel_launch(void* const* d_in, const int* in_sizes, int n_in,
                              void* d_out, int out_size, void* d_ws, size_t ws_size,
                              hipStream_t stream) {
    const float* X   = (const float*)d_in[0];      // (64, 1024, 128) fp32
    float* out       = (float*)d_out;              // scalar fp32
    float* partials  = (float*)d_ws;               // 64 floats of scratch

    gram_batch_kernel<<<NBATCH, 256, 0, stream>>>(X, partials);
    finalize_kernel<<<1, 1, 0, stream>>>(partials, out);
}